// WLConvOptimized_25220047962708
// MI455X (gfx1250) — compile-verified
//
#include <hip/hip_runtime.h>
#include <stdint.h>

// ---------------- WL hash helpers (match reference _mix exactly) ----------------
#define NB_BITS   20
#define NBUCKETS  (1 << NB_BITS)
#define TILE_E    1024           // edges per LDS tile (divides 16M exactly)
#define TPB_TILES 8              // tiles per block in the edge kernel
#define SCAN_TILE 2048           // elements per block in scan kernels

__device__ __forceinline__ uint32_t wlmix(uint32_t v, uint32_t c1, uint32_t c2) {
  v = (v ^ (v >> 16)) * c1;
  v = (v ^ (v >> 13)) * c2;
  return v ^ (v >> 16);
}
__device__ __forceinline__ uint32_t hashA(uint32_t x) { return wlmix(x, 0x045D9F3Bu, 0x045D9F3Bu); }
__device__ __forceinline__ uint32_t hashB(uint32_t x) { return wlmix(x, 0x9E3779B1u, 0x85EBCA77u); }
__device__ __forceinline__ uint32_t sigA(uint32_t ha, uint32_t sa) { return wlmix(ha * 0x27D4EB2Fu + sa, 0xC2B2AE3Du, 0x165667B1u); }
__device__ __forceinline__ uint32_t sigB(uint32_t hb, uint32_t sb) { return wlmix(hb * 0x61C88647u + sb, 0x045D9F3Bu, 0x27D4EB2Fu); }

// LDS byte offset = low 32 bits of the flat address of a __shared__ object
__device__ __forceinline__ uint32_t lds_off(const void* p) {
  return (uint32_t)(uintptr_t)p;
}

// CDNA5 async copy: global -> LDS, 16B per lane, tracked by ASYNCcnt.
__device__ __forceinline__ void async_tile(const int* erow, const int* ecol,
                                           uint32_t ldsR, uint32_t ldsC, uint32_t goff) {
  asm volatile("global_load_async_to_lds_b128 %0, %1, %2"
               :: "v"(ldsR), "v"(goff), "s"(erow) : "memory");
  asm volatile("global_load_async_to_lds_b128 %0, %1, %2"
               :: "v"(ldsC), "v"(goff), "s"(ecol) : "memory");
}

// ---------------- kernels (edge kernel first so it heads the disassembly) ----------------

// Edge scatter-add: streams (row,col) tiles into LDS via async-to-LDS with
// software double-buffering; per-block 64-entry LUT of per-color hashes.
__global__ __launch_bounds__(256) void wl_edge_kernel(
    const int* __restrict__ x, const int* __restrict__ erow, const int* __restrict__ ecol,
    uint32_t* __restrict__ sum_a, uint32_t* __restrict__ sum_b, int nTiles)
{
  __shared__ uint32_t lutA[64];
  __shared__ uint32_t lutB[64];
  __shared__ __align__(16) uint32_t rbuf[2][TILE_E];
  __shared__ __align__(16) uint32_t cbuf[2][TILE_E];

  const int tid   = threadIdx.x;
  const int slice = (tid >> 5) * 128 + (tid & 31) * 4;  // u32 index in tile (8 waves x 128)
  int t0 = blockIdx.x * TPB_TILES;
  if (t0 >= nTiles) return;                             // uniform per block
  int t1 = t0 + TPB_TILES; if (t1 > nTiles) t1 = nTiles;

  const uint32_t ldsR[2] = { lds_off(&rbuf[0][slice]), lds_off(&rbuf[1][slice]) };
  const uint32_t ldsC[2] = { lds_off(&cbuf[0][slice]), lds_off(&cbuf[1][slice]) };

  // kick off tile t0 immediately; compute the LUT while it is in flight
  async_tile(erow, ecol, ldsR[0], ldsC[0], (uint32_t)(t0 * TILE_E + slice) * 4u);

  if (tid < 64) { lutA[tid] = hashA((uint32_t)tid); lutB[tid] = hashB((uint32_t)tid); }

  for (int t = t0; t < t1; ++t) {
    int s = (t - t0) & 1;
    if (t + 1 < t1) {
      async_tile(erow, ecol, ldsR[s ^ 1], ldsC[s ^ 1],
                 (uint32_t)((t + 1) * TILE_E + slice) * 4u);
      asm volatile("s_wait_asynccnt 2" ::: "memory");   // tile t's 2 loads complete
    } else {
      asm volatile("s_wait_asynccnt 0" ::: "memory");
    }
    __syncthreads();
    #pragma unroll
    for (int k = tid; k < TILE_E; k += 256) {
      uint32_t c  = cbuf[s][k];
      uint32_t r  = rbuf[s][k];
      uint32_t xv = (uint32_t)x[r] & 63u;    // 2MB x[] lives in L2
      atomicAdd(&sum_a[c], lutA[xv]);
      atomicAdd(&sum_b[c], lutB[xv]);
    }
    __syncthreads();
  }
}

__global__ void wl_zero(uint32_t* p, int n) {
  int i = blockIdx.x * blockDim.x + threadIdx.x;
  if (i < n) p[i] = 0u;
}

__global__ void wl_edge_tail(const int* x, const int* erow, const int* ecol,
                             uint32_t* sum_a, uint32_t* sum_b, int start, int nE) {
  int e = start + blockIdx.x * blockDim.x + threadIdx.x;
  if (e >= nE) return;
  uint32_t r = (uint32_t)erow[e], c = (uint32_t)ecol[e];
  uint32_t xv = (uint32_t)x[r] & 63u;
  atomicAdd(&sum_a[c], hashA(xv));
  atomicAdd(&sum_b[c], hashB(xv));
}

// Count nodes per top-20-bit bucket of sig_a
__global__ void wl_sig_count(const int* x, const uint32_t* sum_a, uint32_t* cnt, int n) {
  int i = blockIdx.x * blockDim.x + threadIdx.x;
  if (i >= n) return;
  uint32_t sa = sigA(hashA((uint32_t)x[i]), sum_a[i]);
  atomicAdd(&cnt[sa >> (32 - NB_BITS)], 1u);
}

// ----- generic exclusive scan (3 kernels), n <= SCAN_TILE*1024 -----
__global__ __launch_bounds__(256) void scan_reduce(const uint32_t* in, uint32_t* parts, int n) {
  __shared__ uint32_t s[256];
  int base = blockIdx.x * SCAN_TILE + threadIdx.x * 8;
  uint32_t acc = 0;
  #pragma unroll
  for (int j = 0; j < 8; ++j) { int i = base + j; if (i < n) acc += in[i]; }
  s[threadIdx.x] = acc; __syncthreads();
  for (int off = 128; off > 0; off >>= 1) {
    if (threadIdx.x < off) s[threadIdx.x] += s[threadIdx.x + off];
    __syncthreads();
  }
  if (threadIdx.x == 0) parts[blockIdx.x] = s[0];
}

__global__ __launch_bounds__(1024) void scan_top(uint32_t* a, int p) {
  __shared__ uint32_t s[1024];
  int t = threadIdx.x;
  uint32_t v = (t < p) ? a[t] : 0u;
  s[t] = v; __syncthreads();
  for (int off = 1; off < 1024; off <<= 1) {
    uint32_t u = (t >= off) ? s[t - off] : 0u;
    __syncthreads();
    s[t] += u;
    __syncthreads();
  }
  if (t < p) a[t] = s[t] - v;   // exclusive
}

__global__ __launch_bounds__(256) void scan_final(const uint32_t* in, const uint32_t* parts,
                                                  uint32_t* out, int n) {
  __shared__ uint32_t s[256];
  int base = blockIdx.x * SCAN_TILE + threadIdx.x * 8;
  uint32_t v[8]; uint32_t tot = 0;
  #pragma unroll
  for (int j = 0; j < 8; ++j) { int i = base + j; v[j] = (i < n) ? in[i] : 0u; tot += v[j]; }
  s[threadIdx.x] = tot; __syncthreads();
  for (int off = 1; off < 256; off <<= 1) {
    uint32_t u = (threadIdx.x >= off) ? s[threadIdx.x - off] : 0u;
    __syncthreads();
    s[threadIdx.x] += u;
    __syncthreads();
  }
  uint32_t run = parts[blockIdx.x] + (s[threadIdx.x] - tot);
  #pragma unroll
  for (int j = 0; j < 8; ++j) { int i = base + j; if (i < n) out[i] = run; run += v[j]; }
}

// Scatter (key, node) into bucket-ordered arrays (order within bucket arbitrary)
__global__ void wl_scatter(const int* x, const uint32_t* sum_a, const uint32_t* sum_b,
                           uint32_t* offs, uint64_t* skey, uint32_t* sidx, int n) {
  int i = blockIdx.x * blockDim.x + threadIdx.x;
  if (i >= n) return;
  uint32_t xv = (uint32_t)x[i];
  uint32_t sa = sigA(hashA(xv), sum_a[i]);
  uint32_t sb = sigB(hashB(xv), sum_b[i]);
  uint64_t key = ((uint64_t)sa << 32) | (uint64_t)sb;
  uint32_t pos = atomicAdd(&offs[sa >> (32 - NB_BITS)], 1u);
  skey[pos] = key; sidx[pos] = (uint32_t)i;
}

// Tiny per-bucket insertion sort (expected ~0.5 keys/bucket; equal keys are O(1))
__global__ void wl_bucket_sort(const uint32_t* cnt, const uint32_t* offsEnd,
                               uint64_t* skey, uint32_t* sidx, int nb) {
  int b = blockIdx.x * blockDim.x + threadIdx.x;
  if (b >= nb) return;
  int n = (int)cnt[b];
  if (n < 2) return;
  int end = (int)offsEnd[b];       // post-scatter offs[b] == bucket end
  int start = end - n;
  for (int i = start + 1; i < end; ++i) {
    uint64_t k = skey[i]; uint32_t id = sidx[i];
    int j = i - 1;
    while (j >= start && skey[j] > k) { skey[j + 1] = skey[j]; sidx[j + 1] = sidx[j]; --j; }
    skey[j + 1] = k; sidx[j + 1] = id;
  }
}

__global__ void wl_flags(const uint64_t* skey, uint32_t* f, int n) {
  int i = blockIdx.x * blockDim.x + threadIdx.x;
  if (i >= n) return;
  f[i] = (i > 0 && skey[i] != skey[i - 1]) ? 1u : 0u;
}

__global__ void wl_write(const uint32_t* lab, const uint32_t* f, const uint32_t* sidx,
                         float* out, int n) {
  int i = blockIdx.x * blockDim.x + threadIdx.x;
  if (i >= n) return;
  out[sidx[i]] = (float)(lab[i] + f[i]);   // inclusive cumsum of boundary flags
}

// ---------------- launch ----------------
extern "C" void kernel_launch(void* const* d_in, const int* in_sizes, int n_in,
                              void* d_out, int out_size, void* d_ws, size_t ws_size,
                              hipStream_t stream) {
  (void)n_in; (void)out_size; (void)ws_size;
  const int* x  = (const int*)d_in[0];
  const int* ei = (const int*)d_in[1];
  const int  N  = in_sizes[0];
  const int  nE = in_sizes[1] / 2;
  const int* erow = ei;
  const int* ecol = ei + nE;
  float* out = (float*)d_out;

  // workspace layout (u32 units); N is even so u64 arrays stay 8B-aligned
  uint32_t* ws    = (uint32_t*)d_ws;
  uint32_t* sum_a = ws;                               // N
  uint32_t* sum_b = sum_a + N;                        // N
  uint32_t* cnt   = sum_b + N;                        // NBUCKETS
  uint32_t* offs  = cnt + NBUCKETS;                   // NBUCKETS
  uint64_t* skey  = (uint64_t*)(offs + NBUCKETS);     // N u64
  uint32_t* sidx  = (uint32_t*)(skey + N);            // N
  uint32_t* f     = sidx + N;                         // N
  uint32_t* lab   = f + N;                            // N
  uint32_t* parts = lab + N;                          // <=1024

  // 1) zero sum_a, sum_b, cnt (contiguous)
  int z = 2 * N + NBUCKETS;
  wl_zero<<<(z + 255) / 256, 256, 0, stream>>>(ws, z);

  // 2) edge scatter-add (async-to-LDS double-buffered streaming)
  int nTiles = nE / TILE_E;
  if (nTiles > 0) {
    int nBlk = (nTiles + TPB_TILES - 1) / TPB_TILES;
    wl_edge_kernel<<<nBlk, 256, 0, stream>>>(x, erow, ecol, sum_a, sum_b, nTiles);
  }
  int tailStart = nTiles * TILE_E, tail = nE - tailStart;
  if (tail > 0)
    wl_edge_tail<<<(tail + 255) / 256, 256, 0, stream>>>(x, erow, ecol, sum_a, sum_b, tailStart, nE);

  // 3) bucket histogram over top 20 bits of sig_a
  wl_sig_count<<<(N + 255) / 256, 256, 0, stream>>>(x, sum_a, cnt, N);

  // 4) exclusive scan cnt -> offs
  int g1 = (NBUCKETS + SCAN_TILE - 1) / SCAN_TILE;   // 512
  scan_reduce<<<g1, 256, 0, stream>>>(cnt, parts, NBUCKETS);
  scan_top<<<1, 1024, 0, stream>>>(parts, g1);
  scan_final<<<g1, 256, 0, stream>>>(cnt, parts, offs, NBUCKETS);

  // 5) scatter keys into bucket order, 6) sort within buckets
  wl_scatter<<<(N + 255) / 256, 256, 0, stream>>>(x, sum_a, sum_b, offs, skey, sidx, N);
  wl_bucket_sort<<<(NBUCKETS + 255) / 256, 256, 0, stream>>>(cnt, offs, skey, sidx, NBUCKETS);

  // 7) boundary flags, 8) exclusive scan -> labels, 9) scatter to output
  wl_flags<<<(N + 255) / 256, 256, 0, stream>>>(skey, f, N);
  int g2 = (N + SCAN_TILE - 1) / SCAN_TILE;          // 245 for N=500K
  scan_reduce<<<g2, 256, 0, stream>>>(f, parts, N);
  scan_top<<<1, 1024, 0, stream>>>(parts, g2);
  scan_final<<<g2, 256, 0, stream>>>(f, parts, lab, N);
  wl_write<<<(N + 255) / 256, 256, 0, stream>>>(lab, f, sidx, out, N);
}